// DAGNConv_26697516712278
// MI455X (gfx1250) — compile-verified
//
#include <hip/hip_runtime.h>
#include <hip/hip_bf16.h>

#define N_ENT      100000
#define N_EDGES    500000
#define N_REL      200
#define DIM        64
#define HEADS      4
#define HD         256          // HEADS*DIM
#define POW_ITER   5
#define ALPHA      0.1f
#define LEAKY      0.01f
#define EPSV       1e-16f

#define NTILES     (N_ENT / 16)         // 6250 16-entity WMMA tiles

typedef float v2f __attribute__((ext_vector_type(2)));
typedef float v8f __attribute__((ext_vector_type(8)));

// ---------------------------------------------------------------------------
// K1: per-entity attention logits  s_h[n,h], s_t[n,h]
//     proj = ent(16x64) @ W(64x256) via V_WMMA_F32_16X16X4_F32,
//     fused epilogue: tanh -> * att -> reduce over DIM (cross-lane shfl_xor)
// one wave per 16-entity tile; 8 waves / block
// ---------------------------------------------------------------------------
__global__ __launch_bounds__(256) void entity_logits_wmma(
    const float* __restrict__ ent,
    const float* __restrict__ W_h, const float* __restrict__ W_t,
    const float* __restrict__ attH, const float* __restrict__ attT,
    float* __restrict__ s_h, float* __restrict__ s_t)
{
    const int wave = threadIdx.x >> 5;
    const int lane = threadIdx.x & 31;
    const int tile = blockIdx.x * 8 + wave;
    if (tile >= NTILES) return;                 // wave-uniform: EXEC stays all-1s
    const int e0   = tile * 16;
    const int row  = lane & 15;
    const int koff = (lane >> 4) << 1;          // lanes 0-15 -> K {0,1}; 16-31 -> {2,3}

    // Preload A tile (16 entities x 64 K) in WMMA A layout: 16 k-steps of float2
    v2f a[16];
    const float* arow = ent + (size_t)(e0 + row) * DIM;
#pragma unroll
    for (int s = 0; s < 16; ++s) {
        a[s].x = arow[s * 4 + koff];
        a[s].y = arow[s * 4 + koff + 1];
    }

    for (int which = 0; which < 2; ++which) {
        const float* W    = which ? W_t  : W_h;
        const float* av   = which ? attT : attH;
        float*       sout = which ? s_t  : s_h;

        for (int head = 0; head < HEADS; ++head) {
            float psum[8];
#pragma unroll
            for (int v = 0; v < 8; ++v) psum[v] = 0.f;

            for (int ctl = 0; ctl < 4; ++ctl) {             // 4 col-tiles of 16 per head
                const int col = head * DIM + ctl * 16 + (lane & 15);
                v8f acc = {};
#pragma unroll
                for (int s = 0; s < 16; ++s) {              // K = 64, 4 per WMMA
                    const int r0 = s * 4 + koff;
                    v2f b;
                    b.x = W[(size_t)r0       * HD + col];
                    b.y = W[(size_t)(r0 + 1) * HD + col];
                    acc = __builtin_amdgcn_wmma_f32_16x16x4_f32(
                              false, a[s], false, b, (short)0, acc, false, false);
                }
                const float attv = av[col];                 // att[head][d], flat 256
#pragma unroll
                for (int v = 0; v < 8; ++v)
                    psum[v] += tanhf(acc[v]) * attv;
            }
            // reduce over the 16 lanes of each half (N dimension of the tile)
#pragma unroll
            for (int v = 0; v < 8; ++v) {
                float r = psum[v];
                r += __shfl_xor(r, 1, 32);
                r += __shfl_xor(r, 2, 32);
                r += __shfl_xor(r, 4, 32);
                r += __shfl_xor(r, 8, 32);
                if ((lane & 15) == 0) {
                    const int er = e0 + v + ((lane >> 4) ? 8 : 0);
                    sout[er * HEADS + head] = r;
                }
            }
        }
    }
}

// ---------------------------------------------------------------------------
// K1b: per-relation logits s_r[r,h] (tiny: 200 relations) — scalar kernel
// ---------------------------------------------------------------------------
__global__ __launch_bounds__(256) void relation_logits(
    const float* __restrict__ rel, const float* __restrict__ W_r,
    const float* __restrict__ attR, float* __restrict__ s_r)
{
    __shared__ float red[HD];
    const int r = blockIdx.x;
    const int j = threadIdx.x;                  // 0..255 = output column
    const float* rv = rel + (size_t)r * DIM;
    float acc = 0.f;
#pragma unroll 8
    for (int k = 0; k < DIM; ++k)
        acc += rv[k] * W_r[(size_t)k * HD + j];
    const int head = j >> 6;
    const int d    = j & 63;
    red[j] = tanhf(acc) * attR[head * DIM + d];
    __syncthreads();
    for (int off = 32; off > 0; off >>= 1) {
        if (d < off) red[j] += red[j + off];
        __syncthreads();
    }
    if (d == 0) s_r[r * HEADS + head] = red[j];
}

// ---------------------------------------------------------------------------
// Segment softmax over destination (head_ids)
// ---------------------------------------------------------------------------
__global__ void init_segments(unsigned* __restrict__ segmx, float* __restrict__ segsm)
{
    const int i = blockIdx.x * blockDim.x + threadIdx.x;
    if (i >= N_ENT * HEADS) return;
    segmx[i] = 0u;          // ordered-key representation of -inf
    segsm[i] = 0.f;
}

__global__ void edge_scores(
    const int* __restrict__ hid, const int* __restrict__ tid,
    const int* __restrict__ etyp,
    const float* __restrict__ s_h, const float* __restrict__ s_t,
    const float* __restrict__ s_r, float* __restrict__ scores)
{
    const int i = blockIdx.x * blockDim.x + threadIdx.x;
    if (i >= N_EDGES * HEADS) return;
    const int e = i >> 2;
    const int h = i & 3;
    const float x = s_h[hid[e] * HEADS + h] + s_t[tid[e] * HEADS + h]
                  + s_r[etyp[e] * HEADS + h];
    scores[i] = x > 0.f ? x : LEAKY * x;
}

__device__ __forceinline__ unsigned f2key(float f) {
    unsigned u = __float_as_uint(f);
    return (u & 0x80000000u) ? ~u : (u | 0x80000000u);
}
__device__ __forceinline__ float key2f(unsigned k) {
    unsigned u = (k & 0x80000000u) ? (k & 0x7FFFFFFFu) : ~k;
    return __uint_as_float(u);
}

__global__ void seg_max_kernel(const int* __restrict__ hid,
                               const float* __restrict__ scores,
                               unsigned* __restrict__ segmx)
{
    const int i = blockIdx.x * blockDim.x + threadIdx.x;
    if (i >= N_EDGES * HEADS) return;
    const int e = i >> 2, h = i & 3;
    atomicMax(&segmx[hid[e] * HEADS + h], f2key(scores[i]));
}

__global__ void seg_exp_sum(const int* __restrict__ hid,
                            const float* __restrict__ scores,
                            const unsigned* __restrict__ segmx,
                            float* __restrict__ attw, float* __restrict__ segsm)
{
    const int i = blockIdx.x * blockDim.x + threadIdx.x;
    if (i >= N_EDGES * HEADS) return;
    const int e = i >> 2, h = i & 3;
    const float m = key2f(segmx[hid[e] * HEADS + h]);
    const float ev = __expf(scores[i] - m);
    attw[i] = ev;
    atomicAdd(&segsm[hid[e] * HEADS + h], ev);
}

__global__ void seg_normalize(const int* __restrict__ hid,
                              const float* __restrict__ segsm,
                              float* __restrict__ attw)
{
    const int i = blockIdx.x * blockDim.x + threadIdx.x;
    if (i >= N_EDGES * HEADS) return;
    const int e = i >> 2, h = i & 3;
    attw[i] = (1.0f - ALPHA) * attw[i] / (segsm[hid[e] * HEADS + h] + EPSV);
}

// ---------------------------------------------------------------------------
// Power iteration: Z fits in L2 (102.4 MB < 192 MB) -> gather/scatter at L2 BW
// ---------------------------------------------------------------------------
__global__ void init_Z_broadcast(const float* __restrict__ ent, float* __restrict__ Z)
{
    const int i = blockIdx.x * blockDim.x + threadIdx.x;   // over N_ENT*HEADS*DIM
    if (i >= N_ENT * HEADS * DIM) return;
    const int d = i & 63;
    const int n = i >> 8;                                   // / (HEADS*DIM)
    Z[i] = ent[n * DIM + d];
}

__global__ void init_Z_alpha(const float* __restrict__ ent, float* __restrict__ Z)
{
    const int i = blockIdx.x * blockDim.x + threadIdx.x;
    if (i >= N_ENT * HEADS * DIM) return;
    const int d = i & 63;
    const int n = i >> 8;
    Z[i] = ALPHA * ent[n * DIM + d];
}

__global__ void spmm_edges(const int* __restrict__ hid, const int* __restrict__ tid,
                           const float* __restrict__ attw,
                           const float* __restrict__ Zold, float* __restrict__ Znew)
{
    const int i = blockIdx.x * blockDim.x + threadIdx.x;   // N_EDGES*HEADS*16 threads
    if (i >= N_EDGES * HEADS * 16) return;
    const int chunk = i & 15;           // 16 chunks of float4 over DIM
    const int eh    = i >> 4;
    const int h     = eh & 3;
    const int e     = eh >> 2;
    const float a   = attw[e * HEADS + h];
    const float4 z  = *(const float4*)(Zold + (size_t)tid[e] * HD + h * DIM + chunk * 4);
    float* dst = Znew + (size_t)hid[e] * HD + h * DIM + chunk * 4;
    atomicAdd(dst + 0, a * z.x);
    atomicAdd(dst + 1, a * z.y);
    atomicAdd(dst + 2, a * z.z);
    atomicAdd(dst + 3, a * z.w);
}

// ---------------------------------------------------------------------------
// K8: out = Z(100K x 256) @ W_o(256 x 64)  via V_WMMA_F32_16X16X4_F32
// one wave per 16-entity tile, 16x64 output, K=256 in 4-wide steps
// ---------------------------------------------------------------------------
__global__ __launch_bounds__(256) void out_gemm_wmma(
    const float* __restrict__ Z, const float* __restrict__ W_o,
    float* __restrict__ out)
{
    const int wave = threadIdx.x >> 5;
    const int lane = threadIdx.x & 31;
    const int tile = blockIdx.x * 8 + wave;
    if (tile >= NTILES) return;                 // wave-uniform
    const int e0   = tile * 16;
    const int row  = lane & 15;
    const int koff = (lane >> 4) << 1;

    v8f acc[4];
#pragma unroll
    for (int ct = 0; ct < 4; ++ct) acc[ct] = (v8f){};

    const float* zrow = Z + (size_t)(e0 + row) * HD;
    for (int s = 0; s < 64; ++s) {              // K = 256
        const int k0 = s * 4 + koff;
        v2f a;
        a.x = zrow[k0];
        a.y = zrow[k0 + 1];
#pragma unroll
        for (int ct = 0; ct < 4; ++ct) {
            const int col = ct * 16 + row;
            v2f b;
            b.x = W_o[(size_t)k0       * DIM + col];
            b.y = W_o[(size_t)(k0 + 1) * DIM + col];
            acc[ct] = __builtin_amdgcn_wmma_f32_16x16x4_f32(
                          false, a, false, b, (short)0, acc[ct], false, false);
        }
    }

    const int half = (lane >> 4) * 8;
#pragma unroll
    for (int ct = 0; ct < 4; ++ct)
#pragma unroll
        for (int v = 0; v < 8; ++v)
            out[(size_t)(e0 + v + half) * DIM + ct * 16 + (lane & 15)] = acc[ct][v];
}

// ---------------------------------------------------------------------------
extern "C" void kernel_launch(void* const* d_in, const int* in_sizes, int n_in,
                              void* d_out, int out_size, void* d_ws, size_t ws_size,
                              hipStream_t stream)
{
    const float* ent  = (const float*)d_in[0];
    const float* rel  = (const float*)d_in[1];
    const float* W_h  = (const float*)d_in[2];
    const float* W_t  = (const float*)d_in[3];
    const float* W_r  = (const float*)d_in[4];
    const float* attH = (const float*)d_in[5];
    const float* attT = (const float*)d_in[6];
    const float* attR = (const float*)d_in[7];
    const float* W_o  = (const float*)d_in[8];
    const int*   eidx = (const int*)d_in[9];
    const int*   etyp = (const int*)d_in[10];
    const int*   hid  = eidx;                   // edge_index[0]
    const int*   tidp = eidx + N_EDGES;         // edge_index[1]

    // workspace layout (floats)
    float* ws    = (float*)d_ws;
    float* s_h   = ws;                                  // N_ENT*HEADS
    float* s_t   = s_h + (size_t)N_ENT * HEADS;
    float* s_r   = s_t + (size_t)N_ENT * HEADS;         // N_REL*HEADS
    float* segmx = s_r + (size_t)N_REL * HEADS;         // N_ENT*HEADS (uint view)
    float* segsm = segmx + (size_t)N_ENT * HEADS;
    float* attw  = segsm + (size_t)N_ENT * HEADS;       // N_EDGES*HEADS
    float* Zp    = attw + (size_t)N_EDGES * HEADS;      // N_ENT*HD
    float* Zq    = Zp + (size_t)N_ENT * HD;

    const int TB = 256;
    const int tile_blocks = (NTILES + 7) / 8;                       // 782
    const int eh_blocks   = (N_EDGES * HEADS + TB - 1) / TB;        // 7813
    const int seg_blocks  = (N_ENT * HEADS + TB - 1) / TB;          // 1563
    const int z_blocks    = (N_ENT * HEADS * DIM + TB - 1) / TB;    // 100000
    const int spmm_blocks = (N_EDGES * HEADS * 16 + TB - 1) / TB;   // 125000

    // 1) per-entity / per-relation attention logits (WMMA + fused epilogue)
    entity_logits_wmma<<<tile_blocks, TB, 0, stream>>>(ent, W_h, W_t, attH, attT, s_h, s_t);
    relation_logits<<<N_REL, TB, 0, stream>>>(rel, W_r, attR, s_r);

    // 2) edge scores + segment softmax
    init_segments<<<seg_blocks, TB, 0, stream>>>((unsigned*)segmx, segsm);
    edge_scores<<<eh_blocks, TB, 0, stream>>>(hid, tidp, etyp, s_h, s_t, s_r, attw);
    // attw currently holds raw scores; compute max, then exp/sum in-place
    seg_max_kernel<<<eh_blocks, TB, 0, stream>>>(hid, attw, (unsigned*)segmx);
    seg_exp_sum<<<eh_blocks, TB, 0, stream>>>(hid, attw, (const unsigned*)segmx, attw, segsm);
    seg_normalize<<<eh_blocks, TB, 0, stream>>>(hid, segsm, attw);

    // 3) power iteration (Z resident in L2)
    init_Z_broadcast<<<z_blocks, TB, 0, stream>>>(ent, Zp);
    float* Zc = Zp;
    float* Zn = Zq;
    for (int it = 0; it < POW_ITER; ++it) {
        init_Z_alpha<<<z_blocks, TB, 0, stream>>>(ent, Zn);
        spmm_edges<<<spmm_blocks, TB, 0, stream>>>(hid, tidp, attw, Zc, Zn);
        float* t = Zc; Zc = Zn; Zn = t;
    }

    // 4) output GEMM (WMMA)
    out_gemm_wmma<<<tile_blocks, TB, 0, stream>>>(Zc, W_o, (float*)d_out);
}